// LCN_2499670966833
// MI455X (gfx1250) — compile-verified
//
#include <hip/hip_runtime.h>

// Local contrast normalization, (16,1,1024,1024) f32, 11x11 box, reflect pad.
// Memory-bound: ~192 MB min traffic -> ~8.3 us at 23.3 TB/s. Strategy:
// single fused kernel, 16x128 output tile per 256-thread block, raw halo tile
// + horizontal box sums (Sx, Sx^2) in LDS, vertical box sum done as a banded
// matrix multiply D = V(16x28) * T(28x16) on V_WMMA_F32_16X16X4_F32 (f32 end
// to end; 7 K-chunks, band matrix V generated in registers).

typedef __attribute__((ext_vector_type(2))) float v2f;
typedef __attribute__((ext_vector_type(8))) float v8f;

#define IMG_W 1024
#define IMG_H 1024
#define N_IMG 16
#define RAD 5
#define HALO (2 * RAD)        // 10
#define TW 128                // output cols per block
#define THt 16                // output rows per block
#define XW (TW + HALO)        // 138 raw cols
#define XWP 140               // padded raw stride
#define XH (THt + HALO)       // 26 raw rows
#define KROWS 28              // 26 rounded up to multiple of 4 (WMMA K)
#define HSW 136               // hsum stride: 2*136 mod 64 == 16 -> B-frag
                              // half-waves hit disjoint bank ranges

__global__ __launch_bounds__(256) void lcn_wmma_kernel(
    const float* __restrict__ data,
    float* __restrict__ out_norm,
    float* __restrict__ out_std)
{
    __shared__ float X[XH * XWP];        // raw tile with halo
    __shared__ float HS[KROWS * HSW];    // horizontal box sum of x
    __shared__ float HQ[KROWS * HSW];    // horizontal box sum of x^2

    const int tid = threadIdx.x;
    const int img = blockIdx.z;
    const int r0 = blockIdx.y * THt;
    const int c0 = blockIdx.x * TW;
    const float* __restrict__ src = data + (size_t)img * (IMG_W * IMG_H);

    // ---- Stage 1: cooperative load of (26 x 138) halo tile, reflect pad ----
    for (int idx = tid; idx < XH * XWP; idx += 256) {
        int rr = idx / XWP;
        int cc = idx - rr * XWP;
        if (cc < XW) {
            int gr = r0 - RAD + rr;
            gr = gr < 0 ? -gr : gr;
            gr = gr > (IMG_H - 1) ? (2 * IMG_H - 2 - gr) : gr;
            int gc = c0 - RAD + cc;
            gc = gc < 0 ? -gc : gc;
            gc = gc > (IMG_W - 1) ? (2 * IMG_W - 2 - gc) : gc;
            X[idx] = src[gr * IMG_W + gc];
        }
    }
    __syncthreads();

    // ---- Stage 2: horizontal 11-wide running sums (x and x^2) ----
    // Rows 26,27 are K-padding: write exact zeros (WMMA propagates NaN even
    // through a zero A element, so B padding must be real zeros).
    for (int idx = tid; idx < KROWS * TW; idx += 256) {
        int row = idx >> 7;          // TW == 128
        int col = idx & (TW - 1);
        float s = 0.f, q = 0.f;
        if (row < XH) {
            const float* xr = &X[row * XWP + col];
#pragma unroll
            for (int d = 0; d <= 2 * RAD; ++d) {
                float v = xr[d];
                s += v;
                q += v * v;
            }
        }
        HS[row * HSW + col] = s;
        HQ[row * HSW + col] = q;
    }
    __syncthreads();

    // ---- Stage 3: vertical band-matrix multiply on WMMA f32 16x16x4 ----
    // D[m][n] = sum_k V[m][k] * T[k][n], V[m][k] = 1 iff m <= k <= m+10.
    // Wave w owns 16-column tile w. All 8 waves full, EXEC all ones.
    const int lane = tid & 31;
    const int hi = lane >> 4;       // lane half: 0 or 1
    const int n = lane & 15;        // B/D column, and A row (m == lane&15)
    const int ct = tid >> 5;        // column tile = wave id (0..7)
    const int cbase = ct * 16 + n;

    v8f accs = {0.f, 0.f, 0.f, 0.f, 0.f, 0.f, 0.f, 0.f};
    v8f accq = {0.f, 0.f, 0.f, 0.f, 0.f, 0.f, 0.f, 0.f};

#pragma unroll
    for (int kc = 0; kc < 7; ++kc) {
        const int kb = kc * 4 + hi * 2;   // this lane-half's K base
        v2f a, bs, bq;
        // A fragment: banded ones matrix, built in registers.
        a.x = (kb     >= n && kb     <= n + 2 * RAD) ? 1.f : 0.f;
        a.y = (kb + 1 >= n && kb + 1 <= n + 2 * RAD) ? 1.f : 0.f;
        // B fragments from LDS (row kb / kb+1 for this half, col = cbase).
        bs.x = HS[kb * HSW + cbase];
        bs.y = HS[(kb + 1) * HSW + cbase];
        bq.x = HQ[kb * HSW + cbase];
        bq.y = HQ[(kb + 1) * HSW + cbase];
        // (neg_a, A, neg_b, B, c_mod, C, reuse_a, reuse_b)
        accs = __builtin_amdgcn_wmma_f32_16x16x4_f32(
            false, a, false, bs, (short)0, accs, false, false);
        accq = __builtin_amdgcn_wmma_f32_16x16x4_f32(
            false, a, false, bq, (short)0, accq, false, false);
    }

    // ---- Stage 4: epilogue. D VGPR v holds (row v + 8*hi, col n). ----
    const float inv_n = 1.0f / 121.0f;
    const size_t base = (size_t)img * (IMG_W * IMG_H);
#pragma unroll
    for (int v = 0; v < 8; ++v) {
        const int mr = v + hi * 8;                  // tile output row 0..15
        const int grow = r0 + mr;
        const int gcol = c0 + ct * 16 + n;
        const float avg = accs[v] * inv_n;
        float var = fmaf(-avg, avg, accq[v] * inv_n) + 1e-6f;
        var = fmaxf(var, 0.f);
        const float sd = sqrtf(var) + 0.05f;
        const float x = X[(mr + RAD) * XWP + (ct * 16 + n + RAD)];
        const size_t o = base + (size_t)grow * IMG_W + gcol;
        out_norm[o] = (x - avg) / sd;
        out_std[o] = sd;
    }
}

extern "C" void kernel_launch(void* const* d_in, const int* in_sizes, int n_in,
                              void* d_out, int out_size, void* d_ws, size_t ws_size,
                              hipStream_t stream) {
    (void)in_sizes; (void)n_in; (void)out_size; (void)d_ws; (void)ws_size;
    const float* data = (const float*)d_in[0];
    float* out = (float*)d_out;
    float* out_norm = out;                                   // first output
    float* out_std = out + (size_t)N_IMG * IMG_W * IMG_H;    // second output
    dim3 grid(IMG_W / TW, IMG_H / THt, N_IMG);               // (8, 64, 16)
    lcn_wmma_kernel<<<grid, 256, 0, stream>>>(data, out_norm, out_std);
}